// DifferentiableIBS_76055280877726
// MI455X (gfx1250) — compile-verified
//
#include <hip/hip_runtime.h>
#include <hip/hip_bf16.h>
#include <math.h>

// ---- problem constants (match reference setup_inputs) ----
#define BATCHES   4
#define NOBJ      16384
#define NHAND     8192
#define NQ        1024
#define MAX_ITER  40
#define TOL       0.0001f
#define IBS_EPS   1e-10f

#define BLOCKS_PER_BATCH 8
#define QUERIES_PER_BLOCK 128   // 8 waves * 16 queries
#define THREADS 256

typedef __attribute__((ext_vector_type(2))) float v2f;
typedef __attribute__((ext_vector_type(8))) float v8f;
typedef __attribute__((ext_vector_type(4))) unsigned int v4u;
typedef __attribute__((ext_vector_type(8))) int v8i;
typedef __attribute__((ext_vector_type(4))) int v4i;

// LDS layout (floats)
#define HAND_OFF 0
#define OBJ_OFF  (NHAND * 3)                 // 24576
#define SUM_OFF  (OBJ_OFF + NOBJ * 3)        // 73728  (6 used, 8 reserved)
#define SCR_OFF  (SUM_OFF + 8)               // 73736  (8 waves * 32 ints)
#define SMEM_FLOATS (SCR_OFF + 8 * 32)       // 73992 floats = 295968 bytes <= 320KB WGP LDS

__device__ __forceinline__ v8f wmma_16x16x4_f32(v2f a, v2f b) {
  v8f c = {};
#if __has_builtin(__builtin_amdgcn_wmma_f32_16x16x4_f32)
  // 8-arg pattern: (neg_a, A, neg_b, B, c_mod, C, reuse_a, reuse_b)
  return __builtin_amdgcn_wmma_f32_16x16x4_f32(false, a, false, b, (short)0, c,
                                               false, false);
#else
  v8f d;
  asm volatile("v_wmma_f32_16x16x4_f32 %0, %1, %2, 0"
               : "=v"(d)
               : "v"(a), "v"(b));
  (void)c;
  return d;
#endif
}

#if __has_builtin(__builtin_amdgcn_tensor_load_to_lds)
#define HAVE_TDM 1
// One-shot TDM DMA: nfloats contiguous f32 from global -> LDS, as a 1-row 2D tile.
// D# layout per CDNA5 ISA ch.8 (group0 128b, group1 256b, groups 2/3 zero).
// This toolchain exposes the 6-arg builtin:
//   (uint32x4 g0, int32x8 g1, int32x4 g2, int32x4 g3, int32x8 extra, i32 cpol)
__device__ __forceinline__ void tdm_load_row(const float* gsrc,
                                             unsigned int lds_byte_off,
                                             unsigned int nfloats) {
  unsigned long long ga = (unsigned long long)(uintptr_t)gsrc;
  v4u g0;
  g0.x = 1u;                                   // count=1, is_restore=0, gather off
  g0.y = lds_byte_off;                         // lds_addr  [63:32]
  g0.z = (unsigned int)(ga & 0xFFFFFFFFu);     // global_addr[31:0]
  g0.w = (unsigned int)((ga >> 32) & 0x01FFFFFFu) | (2u << 30);  // addr[56:32] | type=2
  v8i g1;
  g1[0] = (int)(2u << 16);                     // wg_mask=0, data_size=4B, no flags
  g1[1] = (int)((nfloats & 0xFFFFu) << 16);    // abar_addr=0 | tensor_dim0[15:0]
  g1[2] = (int)((nfloats >> 16) | (1u << 16)); // tensor_dim0[31:16] | tensor_dim1=1
  g1[3] = (int)((nfloats & 0xFFFFu) << 16);    // tensor_dim1 hi=0 | tile_dim0
  g1[4] = 1;                                   // tile_dim1=1, tile_dim2=0
  g1[5] = (int)nfloats;                        // tensor_dim0_stride[31:0]
  g1[6] = 0;                                   // stride0 hi | tensor_dim1_stride lo
  g1[7] = 0;                                   // tensor_dim1_stride hi
  v4i gz4 = {};
  v8i gz8 = {};
  __builtin_amdgcn_tensor_load_to_lds(g0, g1, gz4, gz4, gz8, 0);
}
#else
#define HAVE_TDM 0
#endif

// Brute-force 1-NN over N targets stored in LDS as packed float3.
// A holds 16 augmented query rows [qx,qy,qz,1]; B cols are [-2tx,-2ty,-2tz,t2],
// so D = d^2 - |q|^2 and argmin_cols(D) == argmin d^2.
// Writes the winning target index for each of the wave's 16 queries to scr[0..15].
template <int N>
__device__ __forceinline__ void nnsearch(const float* __restrict__ t, v2f a,
                                         bool hi, int col, int lane, int* scr) {
  float minv[8];
  int mini[8];
#pragma unroll
  for (int r = 0; r < 8; ++r) { minv[r] = 3.4e38f; mini[r] = 0; }

  for (int n0 = 0; n0 < N; n0 += 16) {
    const float* tp = t + (n0 + col) * 3;
    float tx = tp[0], ty = tp[1], tz = tp[2];
    float t2 = tx * tx + ty * ty + tz * tz;
    v2f b;
    b.x = hi ? (-2.0f * tz) : (-2.0f * tx);   // K=2 : K=0
    b.y = hi ? t2 : (-2.0f * ty);             // K=3 : K=1
    v8f d = wmma_16x16x4_f32(a, b);
#pragma unroll
    for (int r = 0; r < 8; ++r) {
      bool lt = d[r] < minv[r];
      minv[r] = lt ? d[r] : minv[r];
      mini[r] = lt ? n0 : mini[r];
    }
  }
  // materialize true target index (tile base + this lane's column)
#pragma unroll
  for (int r = 0; r < 8; ++r) mini[r] += col;

  // reduce across the 16 lanes of each half (rows 0-7 low half, 8-15 high half)
#pragma unroll
  for (int s = 1; s < 16; s <<= 1) {
#pragma unroll
    for (int r = 0; r < 8; ++r) {
      float ov = __shfl_xor(minv[r], s, 32);
      int oi = __shfl_xor(mini[r], s, 32);
      if (ov < minv[r] || (ov == minv[r] && oi < mini[r])) {
        minv[r] = ov;
        mini[r] = oi;
      }
    }
  }
  if ((lane & 15) == 0) {
    int base = (lane >> 4) * 8;
#pragma unroll
    for (int r = 0; r < 8; ++r) scr[base + r] = mini[r];
  }
}

__global__ void __launch_bounds__(THREADS)
ibs_kernel(const float* __restrict__ obj, const float* __restrict__ hand,
           const float* __restrict__ uvw, float* __restrict__ out) {
  extern __shared__ float smem[];
  float* sHand = smem + HAND_OFF;
  float* sObj = smem + OBJ_OFF;
  float* sSum = smem + SUM_OFF;
  int* sScr = (int*)(smem + SCR_OFF);

  const int tid = threadIdx.x;
  const int lane = tid & 31;
  const int wave = tid >> 5;
  const int batch = blockIdx.x / BLOCKS_PER_BATCH;
  const int col = lane & 15;
  const bool hi = lane >= 16;
  const int q = (blockIdx.x % BLOCKS_PER_BATCH) * QUERIES_PER_BLOCK + wave * 16 + col;

  if (tid < 8) sSum[tid] = 0.0f;

  const float* gh = hand + (size_t)batch * NHAND * 3;
  const float* go = obj + (size_t)batch * NOBJ * 3;

  // ---- stage both clouds into LDS ----
#if HAVE_TDM
  if (wave == 0) {
    // LDS aperture address: low 32 bits are the LDS byte offset (ISA 10.2)
    tdm_load_row(gh, (unsigned int)(uintptr_t)sHand, NHAND * 3);
    tdm_load_row(go, (unsigned int)(uintptr_t)sObj, NOBJ * 3);
    __builtin_amdgcn_s_wait_tensorcnt(0);
  }
#else
  for (int i = tid; i < NHAND * 3; i += THREADS) sHand[i] = gh[i];
  for (int i = tid; i < NOBJ * 3; i += THREADS) sObj[i] = go[i];
#endif
  __syncthreads();

  // ---- per-batch means (from LDS) ----
  float shx = 0.f, shy = 0.f, shz = 0.f;
  for (int i = tid; i < NHAND; i += THREADS) {
    shx += sHand[i * 3 + 0]; shy += sHand[i * 3 + 1]; shz += sHand[i * 3 + 2];
  }
  float sox = 0.f, soy = 0.f, soz = 0.f;
  for (int i = tid; i < NOBJ; i += THREADS) {
    sox += sObj[i * 3 + 0]; soy += sObj[i * 3 + 1]; soz += sObj[i * 3 + 2];
  }
#pragma unroll
  for (int s = 16; s >= 1; s >>= 1) {
    shx += __shfl_down(shx, s, 32); shy += __shfl_down(shy, s, 32);
    shz += __shfl_down(shz, s, 32);
    sox += __shfl_down(sox, s, 32); soy += __shfl_down(soy, s, 32);
    soz += __shfl_down(soz, s, 32);
  }
  if (lane == 0) {
    atomicAdd(&sSum[0], shx); atomicAdd(&sSum[1], shy); atomicAdd(&sSum[2], shz);
    atomicAdd(&sSum[3], sox); atomicAdd(&sSum[4], soy); atomicAdd(&sSum[5], soz);
  }
  __syncthreads();

  const float hcx = sSum[0] * (1.0f / NHAND), hcy = sSum[1] * (1.0f / NHAND),
              hcz = sSum[2] * (1.0f / NHAND);
  const float ocx = sSum[3] * (1.0f / NOBJ), ocy = sSum[4] * (1.0f / NOBJ),
              ocz = sSum[5] * (1.0f / NOBJ);
  const float cx = 0.5f * (hcx + ocx), cy = 0.5f * (hcy + ocy),
              cz = 0.5f * (hcz + ocz);
  const float dxc = hcx - ocx, dyc = hcy - ocy, dzc = hcz - ocz;
  const float rv = 0.8f * sqrtf(dxc * dxc + dyc * dyc + dzc * dzc) + 0.05f;

  // ---- initial point for this lane's query (mirrored across both halves) ----
  const float* gu = uvw + ((size_t)batch * NQ + q) * 3;
  float u = gu[0], v = gu[1], w = gu[2];
  float r = rv * cbrtf(u);
  float theta = acosf(2.0f * v - 1.0f);
  float phi = 6.283185307179586f * w;
  float st = sinf(theta);
  float px = cx + r * st * cosf(phi);
  float py = cy + r * st * sinf(phi);
  float pz = cz + r * cosf(theta);

  int* scrH = sScr + wave * 32;
  int* scrO = scrH + 16;

  for (int it = 0; it < MAX_ITER; ++it) {
    // augmented A row [qx,qy,qz,1]: low half holds K=0,1; high half K=2,3
    v2f a;
    a.x = hi ? pz : px;
    a.y = hi ? 1.0f : py;

    nnsearch<NHAND>(sHand, a, hi, col, lane, scrH);
    nnsearch<NOBJ>(sObj, a, hi, col, lane, scrO);
    __syncthreads();

    int ih = scrH[col];
    int io = scrO[col];

    float hx = sHand[ih * 3 + 0] - px, hy = sHand[ih * 3 + 1] - py,
          hz = sHand[ih * 3 + 2] - pz;
    float ox = sObj[io * 3 + 0] - px, oy = sObj[io * 3 + 1] - py,
          oz = sObj[io * 3 + 2] - pz;
    float hd = sqrtf(hx * hx + hy * hy + hz * hz);
    float od = sqrtf(ox * ox + oy * oy + oz * oz);
    float invh = 1.0f / (hd + IBS_EPS);
    float invo = 1.0f / (od + IBS_EPS);
    float hnx = hx * invh, hny = hy * invh, hnz = hz * invh;
    float onx = ox * invo, ony = oy * invo, onz = oz * invo;

    float sgn = hd - od;
    float msk = (fabsf(sgn) >= TOL) ? 1.0f : 0.0f;
    float dotn = hnx * onx + hny * ony + hnz * onz;
    bool pos = sgn >= 0.0f;
    float denom = pos ? (hd - od * dotn) : (od - hd * dotn);
    float weight = 0.5f * (hd + od) / (denom + IBS_EPS);
    float coef = weight * fabsf(sgn) * msk;
    px += coef * (pos ? hnx : onx);
    py += coef * (pos ? hny : ony);
    pz += coef * (pos ? hnz : onz);

    __syncthreads();  // protect per-wave scratch against next iteration's writes
  }

  if (lane < 16) {
    float* po = out + ((size_t)batch * NQ + q) * 3;
    po[0] = px; po[1] = py; po[2] = pz;
  }
}

extern "C" void kernel_launch(void* const* d_in, const int* in_sizes, int n_in,
                              void* d_out, int out_size, void* d_ws,
                              size_t ws_size, hipStream_t stream) {
  (void)in_sizes; (void)n_in; (void)d_ws; (void)ws_size; (void)out_size;
  const float* obj = (const float*)d_in[0];
  const float* hand = (const float*)d_in[1];
  const float* uvw = (const float*)d_in[2];
  float* out = (float*)d_out;

  dim3 grid(BATCHES * BLOCKS_PER_BATCH);  // 32 blocks, one batch-slice each
  dim3 block(THREADS);
  size_t shmem = (size_t)SMEM_FLOATS * sizeof(float);  // ~289 KB dynamic LDS
  hipLaunchKernelGGL(ibs_kernel, grid, block, shmem, stream, obj, hand, uvw, out);
}